// CBFLayer_48739288875026
// MI455X (gfx1250) — compile-verified
//
#include <hip/hip_runtime.h>

// ---------------------------------------------------------------------------
// Batched CBF-QP safety filter, fused single kernel for gfx1250 (MI455X).
//
// Per 16-batch tile (one 128-thread / 4-wave workgroup):
//   S1: QX[b, m, i] = (Qc[m] x_b)_i  via V_WMMA_F32_16X16X4_F32
//       (X tile is the 16x64 A-matrix, Qc is the shared 64x512 B-matrix)
//   S2: g_b[m] = 0.5 x.QX + cc.x - dc          (VALU, LDS)
//   S3: Ax_b = A[b] x_b                        (VALU, streams 256KB/tile of A)
//   S4: bvec = (QX+cc).Ax + g                  ( = -Lf_h - DELTA*h_x )
//   S5: Lg_b = -(QX+cc) @ B[b]   [8x16]        (VALU, streams B)
//   S6: S = Lg Lg^T (8x8, rows in regs), q1 = bvec - Lg a_des
//   S7: lambda_max(M) by power iteration  (M.v via 8-lane shuffle groups)
//   S8: 250 FISTA iterations fully in registers, M*y collapsed to
//       0.5*S*y1 + 0.05*(y1+y2) coupling  (8 lanes per batch element)
//   S9: a_safe = a_des + 0.5 * Lg^T lam1
// ---------------------------------------------------------------------------

#define XD 64
#define AD 16
#define NC 8
#define DELTA 1.0f
#define QP_ITERS 250
#define POW_ITERS 64
#define TILE 16
#define THREADS 128

typedef float v8f __attribute__((ext_vector_type(8)));
typedef float v2f __attribute__((ext_vector_type(2)));

__launch_bounds__(THREADS)
__global__ void cbf_fused_kernel(const float* __restrict__ a_des,
                                 const float* __restrict__ xg,
                                 const float* __restrict__ Ag,
                                 const float* __restrict__ Bg,
                                 const float* __restrict__ Qcg,
                                 const float* __restrict__ ccg,
                                 const float* __restrict__ dcg,
                                 float* __restrict__ outp)
{
    // ---- LDS (padded against 64-bank conflicts) ---------------------------
    __shared__ float xs [TILE][XD + 1];          // x tile
    __shared__ float QX [TILE][NC * XD + 4];     // Qc[m] x  (516 floats/row)
    __shared__ float ccs[NC][XD];
    __shared__ float dcs[NC];
    __shared__ float Axs[TILE][XD + 4];
    __shared__ float bvs[TILE][NC];
    __shared__ float Lgs[TILE][NC * AD + 4];     // 132 floats/row
    __shared__ float lam1s[TILE][NC];

    const int t    = threadIdx.x;
    const int bb0  = blockIdx.x * TILE;
    const int lane = t & 31;
    const int wave = t >> 5;

    // ---- Stage 0: stage x tile / cc / dc into LDS (coalesced) -------------
    for (int i = t; i < TILE * XD; i += THREADS)
        xs[i >> 6][i & 63] = xg[(size_t)bb0 * XD + i];
    for (int i = t; i < NC * XD; i += THREADS)
        ccs[i >> 6][i & 63] = ccg[i];
    if (t < NC) dcs[t] = dcg[t];
    __syncthreads();

    // ---- Stage 1: QX = X * W  via f32 WMMA  -------------------------------
    // A-frag (16x4 f32): v0/v1 = K0/K1 on lanes 0-15, K2/K3 on lanes 16-31.
    // B-frag (4x16 f32): mirrored; lanes hold N columns.
    {
        const int row = lane & 15;
        const int hi  = lane >> 4;
        float afrag[32];
#pragma unroll
        for (int kk = 0; kk < 16; ++kk) {
            afrag[2 * kk]     = xs[row][4 * kk + 2 * hi];
            afrag[2 * kk + 1] = xs[row][4 * kk + 2 * hi + 1];
        }
#pragma unroll
        for (int mi = 0; mi < 2; ++mi) {
            const int m = wave * 2 + mi;            // wave-uniform
#pragma unroll
            for (int nt = 0; nt < 4; ++nt) {
                const int col = nt * 16 + row;      // output column i
                v8f acc = {0.f, 0.f, 0.f, 0.f, 0.f, 0.f, 0.f, 0.f};
#pragma unroll
                for (int kk = 0; kk < 16; ++kk) {
                    const int j0 = 4 * kk + 2 * hi;
                    v2f av; av.x = afrag[2 * kk]; av.y = afrag[2 * kk + 1];
                    v2f bv;
                    bv.x = Qcg[(m * XD + j0) * XD + col];
                    bv.y = Qcg[(m * XD + j0 + 1) * XD + col];
                    acc = __builtin_amdgcn_wmma_f32_16x16x4_f32(
                        false, av, false, bv, (short)0, acc, false, false);
                }
                // C/D layout: VGPR r -> rows r (lanes 0-15) / r+8 (16-31)
#pragma unroll
                for (int r = 0; r < 8; ++r)
                    QX[hi * 8 + r][m * XD + nt * 16 + row] = acc[r];
            }
        }
    }
    __syncthreads();

    // ---- Stage 2: constraint values g_b[m] --------------------------------
    float gval;
    {
        const int bb = t & 15;      // 516 % 64 == 4 -> distinct banks per bb
        const int m  = t >> 4;
        float s1 = 0.f, s2 = 0.f;
        for (int i = 0; i < XD; ++i) {
            const float xi = xs[bb][i];
            s1 = fmaf(QX[bb][m * XD + i], xi, s1);
            s2 = fmaf(ccs[m][i], xi, s2);
        }
        gval = 0.5f * s1 + s2 - dcs[m];
    }

    // ---- Stage 3: Ax (batched matvec, streams A) --------------------------
    for (int p = 0; p < 8; ++p) {
        const int bb = 2 * p + (t >> 6);
        const int r  = t & 63;
        const float* Arow = Ag + (((size_t)(bb0 + bb)) * XD + r) * XD;
        float s = 0.f;
#pragma unroll
        for (int j = 0; j < XD; j += 4) {
            const float4 a4 = *(const float4*)(Arow + j);
            s = fmaf(a4.x, xs[bb][j],     s);
            s = fmaf(a4.y, xs[bb][j + 1], s);
            s = fmaf(a4.z, xs[bb][j + 2], s);
            s = fmaf(a4.w, xs[bb][j + 3], s);
        }
        Axs[bb][r] = s;
    }
    __syncthreads();

    // ---- Stage 4: bvec = -Lf_h - DELTA*h = (QX+cc).Ax + DELTA*g -----------
    {
        const int bb = t & 15;
        const int m  = t >> 4;
        float s = 0.f;
        for (int i = 0; i < XD; ++i)
            s = fmaf(QX[bb][m * XD + i] + ccs[m][i], Axs[bb][i], s);
        bvs[bb][m] = s + DELTA * gval;
    }

    // ---- Stage 5: Lg = grad_h @ B[b]   [8 x 16] ---------------------------
    {
        const int bb = t >> 3;
        const int ap = t & 7;                 // column pair 2*ap, 2*ap+1
        float acc[16];
#pragma unroll
        for (int z = 0; z < 16; ++z) acc[z] = 0.f;
        const float* Bb = Bg + ((size_t)(bb0 + bb)) * XD * AD;
        for (int i = 0; i < XD; ++i) {
            const float b0 = Bb[i * AD + 2 * ap];
            const float b1 = Bb[i * AD + 2 * ap + 1];
#pragma unroll
            for (int m = 0; m < NC; ++m) {    // QX/ccs reads broadcast
                const float g = -(QX[bb][m * XD + i] + ccs[m][i]);
                acc[2 * m]     = fmaf(g, b0, acc[2 * m]);
                acc[2 * m + 1] = fmaf(g, b1, acc[2 * m + 1]);
            }
        }
#pragma unroll
        for (int m = 0; m < NC; ++m) {
            Lgs[bb][m * AD + 2 * ap]     = acc[2 * m];
            Lgs[bb][m * AD + 2 * ap + 1] = acc[2 * m + 1];
        }
    }
    __syncthreads();

    // ---- Stage 6: S rows (scaled by 0.5) and q1, per (batch,row) thread ---
    const int bb = t >> 3;                    // 8 lanes per batch element
    const int k  = t & 7;
    float LgRow[AD];
#pragma unroll
    for (int a = 0; a < AD; ++a) LgRow[a] = Lgs[bb][k * AD + a];
    float SrowH[NC];
#pragma unroll
    for (int l = 0; l < NC; ++l) {
        float s = 0.f;
#pragma unroll
        for (int a = 0; a < AD; ++a)
            s = fmaf(LgRow[a], Lgs[bb][l * AD + a], s);
        SrowH[l] = 0.5f * s;                  // fold dinv=0.5 into S
    }
    float q1k = bvs[bb][k];
    {
        const float* ad = a_des + ((size_t)(bb0 + bb)) * AD;
#pragma unroll
        for (int a = 0; a < AD; ++a) q1k = fmaf(-LgRow[a], ad[a], q1k);
    }

    // ---- Stage 7: lambda_max(M) via power iteration (8-lane groups) -------
    float v1 = 1.0f, v2 = 1.0f, lmax = 0.1f;
    for (int it = 0; it < POW_ITERS; ++it) {
        float sy = 0.f;
#pragma unroll
        for (int j = 0; j < NC; ++j)
            sy = fmaf(SrowH[j], __shfl(v1, j, 8), sy);
        const float c  = 0.05f * (v1 + v2);   // 1/(2*PEN) coupling
        const float u1 = sy + c;
        const float u2 = c;
        float n2 = fmaf(u1, u1, u2 * u2);
        n2 += __shfl_xor(n2, 1, 8);
        n2 += __shfl_xor(n2, 2, 8);
        n2 += __shfl_xor(n2, 4, 8);
        lmax = sqrtf(n2 + 1e-30f);
        const float inv = 1.0f / lmax;
        v1 = u1 * inv; v2 = u2 * inv;
    }
    const float tstep = 1.0f / (lmax + 1e-6f);

    // ---- Stage 8: FISTA (250 iters, registers + shuffles only) ------------
    float l1 = 0.f, l2 = 0.f, y1 = 0.f, y2 = 0.f, tk = 1.0f;
    for (int it = 0; it < QP_ITERS; ++it) {
        float sy = 0.f;
#pragma unroll
        for (int j = 0; j < NC; ++j)
            sy = fmaf(SrowH[j], __shfl(y1, j, 8), sy);
        const float c   = 0.05f * (y1 + y2);
        const float g1  = sy + c - q1k;
        const float g2  = c;
        const float n1  = fmaxf(fmaf(-tstep, g1, y1), 0.0f);
        const float n2v = fmaxf(fmaf(-tstep, g2, y2), 0.0f);
        const float tk1  = 0.5f * (1.0f + sqrtf(fmaf(4.0f * tk, tk, 1.0f)));
        const float beta = (tk - 1.0f) / tk1;
        y1 = fmaf(beta, n1 - l1, n1);
        y2 = fmaf(beta, n2v - l2, n2v);
        l1 = n1; l2 = n2v; tk = tk1;
    }
    lam1s[bb][k] = l1;
    __syncthreads();

    // ---- Stage 9: a_safe = a_des + 0.5 * Lg^T lam1 (coalesced) ------------
#pragma unroll
    for (int rep = 0; rep < 2; ++rep) {
        const int idx = t + rep * THREADS;
        const int ob  = idx >> 4;
        const int a   = idx & 15;
        float s = a_des[((size_t)(bb0 + ob)) * AD + a];
#pragma unroll
        for (int m = 0; m < NC; ++m)
            s = fmaf(0.5f * lam1s[ob][m], Lgs[ob][m * AD + a], s);
        outp[((size_t)(bb0 + ob)) * AD + a] = s;
    }
}

extern "C" void kernel_launch(void* const* d_in, const int* in_sizes, int n_in,
                              void* d_out, int out_size, void* d_ws, size_t ws_size,
                              hipStream_t stream) {
    (void)in_sizes; (void)n_in; (void)d_ws; (void)ws_size;
    const float* a_des = (const float*)d_in[0];
    const float* x     = (const float*)d_in[1];
    const float* A     = (const float*)d_in[2];
    const float* B     = (const float*)d_in[3];
    const float* Qc    = (const float*)d_in[4];
    const float* cc    = (const float*)d_in[5];
    const float* dc    = (const float*)d_in[6];
    float* out = (float*)d_out;
    const int batch = out_size / AD;          // 16384
    const int grid  = batch / TILE;           // 1024 workgroups
    hipLaunchKernelGGL(cbf_fused_kernel, dim3(grid), dim3(THREADS), 0, stream,
                       a_des, x, A, B, Qc, cc, dc, out);
}